// RPCA_4D_GPU_71494025609648
// MI455X (gfx1250) — compile-verified
//
#include <hip/hip_runtime.h>
#include <math.h>

// ---------------------------------------------------------------------------
// Batched RPCA (inexact ALM, randomized SVD rank 10) for 64 x (512x512) fp32.
// One workgroup (256 threads = 8 wave32) per matrix; 10 ALM iterations inside
// the kernel with workgroup barriers. All GEMMs run on V_WMMA_F32_16X16X4_F32.
// ---------------------------------------------------------------------------

typedef float v2f __attribute__((ext_vector_type(2)));
typedef float v8f __attribute__((ext_vector_type(8)));

#define HH      512
#define WW      512
#define NMAT    64
#define KP      16          // padded rank (WMMA tile)
#define KR      10          // true randomized-SVD rank
#define NTHR    256
#define ELEMS   (HH * WW)   // 262144
#define MAXIT   10
#define TOLF    1e-7f

// D = A(16x4 f32) * B(4x16 f32) + C(16x16 f32)
static __device__ inline v8f wmma4(v2f a, v2f b, v8f c) {
  return __builtin_amdgcn_wmma_f32_16x16x4_f32(
      /*neg_a=*/false, a, /*neg_b=*/false, b,
      /*c_mod=*/(short)0, c, /*reuse_a=*/false, /*reuse_b=*/false);
}

static __device__ inline unsigned pcg(unsigned x) {
  unsigned s = x * 747796405u + 2891336453u;
  unsigned w = ((s >> ((s >> 28u) + 4u)) ^ s) * 277803737u;
  return (w >> 22u) ^ w;
}

static __device__ inline float gauss_hash(unsigned seed) {
  unsigned a = pcg(seed);
  unsigned b = pcg(seed ^ 0x9E3779B9u);
  float u1 = ((float)a + 1.0f) * 2.3283064365386963e-10f;   // (0,1]
  float u2 = (float)b * 2.3283064365386963e-10f;            // [0,1)
  return sqrtf(-2.0f * __logf(u1)) * __cosf(6.28318530718f * u2);
}

static __device__ inline float shrinkf(float x, float tau) {
  float m = fabsf(x) - tau;
  m = fmaxf(m, 0.0f);
  return (x >= 0.0f) ? m : -m;
}

static __device__ inline float block_reduce_sum(float v, float* red) {
  const int t = threadIdx.x;
  red[t] = v;
  __syncthreads();
  for (int s = NTHR / 2; s > 0; s >>= 1) {
    if (t < s) red[t] += red[t + s];
    __syncthreads();
  }
  float r = red[0];
  __syncthreads();
  return r;
}

__global__ __launch_bounds__(NTHR)
void rpca_alm_kernel(const float* __restrict__ Dall,
                     float* __restrict__ out,     // [L (64*512*512) | S (...)]
                     float* __restrict__ ws) {    // [Y (64 MB) | X (64 MB)]
  __shared__ float Pbuf[HH * KP];     // 32 KB: P, later QM = Q*M
  __shared__ float ZQ[HH * KP];       // 32 KB: Z, later Q (CholQR in place)
  __shared__ float Ys[KP * WW];       // 32 KB: Y_small = Q^T X
  __shared__ float Gm[KP * KP];       // Gram scratch
  __shared__ float Mb[KP * KP];       // R^-1  /  U diag(f) U^T
  __shared__ float red[NTHR];
  __shared__ float sc[4];             // mu, 1/mu, lam/mu, normD

  const int b = blockIdx.x;
  const float* Dg = Dall + (size_t)b * ELEMS;
  float* Lg = out + (size_t)b * ELEMS;
  float* Sg = out + (size_t)NMAT * ELEMS + (size_t)b * ELEMS;
  float* Yg = ws + (size_t)b * ELEMS;
  float* Xg = ws + (size_t)NMAT * ELEMS + (size_t)b * ELEMS;

  const int t    = threadIdx.x;
  const int wave = t >> 5;
  const int lane = t & 31;
  const int half = lane >> 4;
  const int l    = lane & 15;

  const float4* D4 = (const float4*)Dg;
  float4* L4 = (float4*)Lg;
  float4* S4 = (float4*)Sg;
  float4* Y4 = (float4*)Yg;
  float4* X4 = (float4*)Xg;

  // ---- scalars: mu = m*n / (4*sum|D|), lam = 1/sqrt(512), normD -----------
  float sAbs = 0.0f, sSq = 0.0f;
  for (int i = t; i < ELEMS / 4; i += NTHR) {
    float4 d = D4[i];
    sAbs += fabsf(d.x) + fabsf(d.y) + fabsf(d.z) + fabsf(d.w);
    sSq  += d.x * d.x + d.y * d.y + d.z * d.z + d.w * d.w;
  }
  float totAbs = block_reduce_sum(sAbs, red);
  float totSq  = block_reduce_sum(sSq, red);
  if (t == 0) {
    float mu = (float)ELEMS / (4.0f * totAbs);
    sc[0] = mu;
    sc[1] = 1.0f / mu;
    sc[2] = (1.0f / sqrtf(512.0f)) / mu;
    sc[3] = sqrtf(totSq);
  }
  __syncthreads();
  const float mu = sc[0], inv_mu = sc[1], lam_mu = sc[2];
  const float inv_normD = 1.0f / sc[3];

  // ---- init: L = 0, S = 0, Y = D / ||D|| ----------------------------------
  for (int i = t; i < ELEMS / 4; i += NTHR) {
    float4 d = D4[i];
    float4 z; z.x = 0.f; z.y = 0.f; z.z = 0.f; z.w = 0.f;
    L4[i] = z; S4[i] = z;
    float4 y;
    y.x = d.x * inv_normD; y.y = d.y * inv_normD;
    y.z = d.z * inv_normD; y.w = d.w * inv_normD;
    Y4[i] = y;
  }
  __threadfence();
  __syncthreads();

  for (int iter = 0; iter < MAXIT; ++iter) {
    // (1) random projection P (512 x 10, padded to 16 with zero cols)
    for (int i = t; i < HH * KP; i += NTHR) {
      int r = i >> 4, c = i & 15;
      float v = 0.0f;
      if (c < KR)
        v = gauss_hash((unsigned)(b * 131071 + iter * 8209 + r * 17 + c) *
                           2654435761u + 12345u);
      Pbuf[i] = v;
    }
    // (2) X = D - S + Y / mu  (streamed to global workspace)
    for (int i = t; i < ELEMS / 4; i += NTHR) {
      float4 d = D4[i], s = S4[i], y = Y4[i];
      float4 x;
      x.x = d.x - s.x + y.x * inv_mu;
      x.y = d.y - s.y + y.y * inv_mu;
      x.z = d.z - s.z + y.z * inv_mu;
      x.w = d.w - s.w + y.w * inv_mu;
      X4[i] = x;
    }
    __threadfence();
    __syncthreads();

    // (3) Z = X @ P   (512x512 * 512x16) -> ZQ in LDS
    for (int ti = 0; ti < 4; ++ti) {
      const int m0 = (wave + ti * 8) * 16;
      v8f acc = {0.f, 0.f, 0.f, 0.f, 0.f, 0.f, 0.f, 0.f};
      const float* arow = Xg + (size_t)(m0 + l) * WW + 2 * half;
      for (int k0 = 0; k0 < WW; k0 += 4) {
        __builtin_prefetch(arow + k0 + 64, 0, 0);
        v2f a;  a.x = arow[k0];  a.y = arow[k0 + 1];
        v2f bb; bb.x = Pbuf[(k0 + 2 * half) * KP + l];
                bb.y = Pbuf[(k0 + 2 * half + 1) * KP + l];
        acc = wmma4(a, bb, acc);
      }
      for (int g = 0; g < 8; ++g)
        ZQ[(m0 + g + 8 * half) * KP + l] = acc[g];
    }
    __syncthreads();

    // (4) Gram G = Z^T Z (16x16; zero-padded cols stay zero)
    {
      const int gi = t >> 4, gj = t & 15;
      float s = 0.0f;
      for (int r = 0; r < HH; ++r)
        s += ZQ[r * KP + gi] * ZQ[r * KP + gj];
      Gm[t] = s;
    }
    __syncthreads();

    // (5) thread 0: Cholesky (G = R^T R) on 10x10 + R^-1 -> Mb (padded 16x16)
    if (t == 0) {
      float A[KR][KR], R[KR][KR], Ri[KR][KR];
      float tr = 0.0f;
      for (int i = 0; i < KR; ++i) tr += Gm[i * KP + i];
      float jit = 1e-6f * (tr / KR) + 1e-20f;
      for (int i = 0; i < KR; ++i)
        for (int j = 0; j < KR; ++j)
          A[i][j] = Gm[i * KP + j] + (i == j ? jit : 0.0f);
      for (int i = 0; i < KR; ++i) {
        float s = A[i][i];
        for (int k = 0; k < i; ++k) s -= R[k][i] * R[k][i];
        R[i][i] = sqrtf(fmaxf(s, 1e-20f));
        for (int j = i + 1; j < KR; ++j) {
          float s2 = A[i][j];
          for (int k = 0; k < i; ++k) s2 -= R[k][i] * R[k][j];
          R[i][j] = s2 / R[i][i];
        }
      }
      for (int i = 0; i < KR; ++i)
        for (int j = 0; j < KR; ++j) Ri[i][j] = 0.0f;
      for (int j = 0; j < KR; ++j) {
        Ri[j][j] = 1.0f / R[j][j];
        for (int i = j - 1; i >= 0; --i) {
          float s3 = 0.0f;
          for (int k = i + 1; k <= j; ++k) s3 += R[i][k] * Ri[k][j];
          Ri[i][j] = -s3 / R[i][i];
        }
      }
      for (int i = 0; i < KP; ++i)
        for (int j = 0; j < KP; ++j)
          Mb[i * KP + j] = (i < KR && j < KR) ? Ri[i][j] : 0.0f;
    }
    __syncthreads();

    // (6) Q = Z @ R^-1 (CholeskyQR, in place in ZQ)
    {
      v8f acc[4];
      for (int ti = 0; ti < 4; ++ti) {
        const int m0 = (wave + ti * 8) * 16;
        v8f c = {0.f, 0.f, 0.f, 0.f, 0.f, 0.f, 0.f, 0.f};
        for (int k0 = 0; k0 < KP; k0 += 4) {
          v2f a;  a.x = ZQ[(m0 + l) * KP + k0 + 2 * half];
                  a.y = ZQ[(m0 + l) * KP + k0 + 2 * half + 1];
          v2f bb; bb.x = Mb[(k0 + 2 * half) * KP + l];
                  bb.y = Mb[(k0 + 2 * half + 1) * KP + l];
          c = wmma4(a, bb, c);
        }
        acc[ti] = c;
      }
      __syncthreads();
      for (int ti = 0; ti < 4; ++ti) {
        const int m0 = (wave + ti * 8) * 16;
        for (int g = 0; g < 8; ++g)
          ZQ[(m0 + g + 8 * half) * KP + l] = acc[ti][g];
      }
    }
    __syncthreads();

    // (7) Y_small = Q^T @ X  (16x512 result in LDS)
    for (int ti = 0; ti < 4; ++ti) {
      const int n0 = (wave + ti * 8) * 16;
      v8f c = {0.f, 0.f, 0.f, 0.f, 0.f, 0.f, 0.f, 0.f};
      for (int k0 = 0; k0 < HH; k0 += 4) {
        __builtin_prefetch(Xg + (size_t)(k0 + 8) * WW + n0 + l, 0, 0);
        v2f a;  a.x = ZQ[(k0 + 2 * half) * KP + l];
                a.y = ZQ[(k0 + 2 * half + 1) * KP + l];
        v2f bb; bb.x = Xg[(size_t)(k0 + 2 * half) * WW + n0 + l];
                bb.y = Xg[(size_t)(k0 + 2 * half + 1) * WW + n0 + l];
        c = wmma4(a, bb, c);
      }
      for (int g = 0; g < 8; ++g)
        Ys[(g + 8 * half) * WW + n0 + l] = c[g];
    }
    __syncthreads();

    // (8) G2 = Y_small Y_small^T (16x16)
    {
      const int gi = t >> 4, gj = t & 15;
      float s = 0.0f;
      for (int n = 0; n < WW; ++n)
        s += Ys[gi * WW + n] * Ys[gj * WW + n];
      Gm[t] = s;
    }
    __syncthreads();

    // (9) thread 0: Jacobi eigen of G2 -> M = U diag(shr(s)/s) U^T -> Mb
    if (t == 0) {
      float A[KR][KR], V[KR][KR];
      for (int i = 0; i < KR; ++i)
        for (int j = 0; j < KR; ++j) {
          A[i][j] = Gm[i * KP + j];
          V[i][j] = (i == j) ? 1.0f : 0.0f;
        }
      for (int sweep = 0; sweep < 10; ++sweep) {
        for (int p = 0; p < KR - 1; ++p)
          for (int q = p + 1; q < KR; ++q) {
            float apq = A[p][q];
            if (fabsf(apq) < 1e-14f) continue;
            float th = 0.5f * (A[q][q] - A[p][p]) / apq;
            float tt = (th >= 0.0f ? 1.0f : -1.0f) /
                       (fabsf(th) + sqrtf(1.0f + th * th));
            float cc = 1.0f / sqrtf(1.0f + tt * tt);
            float ss = tt * cc;
            for (int k = 0; k < KR; ++k) {
              float akp = A[k][p], akq = A[k][q];
              A[k][p] = cc * akp - ss * akq;
              A[k][q] = ss * akp + cc * akq;
            }
            for (int k = 0; k < KR; ++k) {
              float apk = A[p][k], aqk = A[q][k];
              A[p][k] = cc * apk - ss * aqk;
              A[q][k] = ss * apk + cc * aqk;
            }
            for (int k = 0; k < KR; ++k) {
              float vkp = V[k][p], vkq = V[k][q];
              V[k][p] = cc * vkp - ss * vkq;
              V[k][q] = ss * vkp + cc * vkq;
            }
          }
      }
      float f[KR];
      for (int i = 0; i < KR; ++i) {
        float sv = sqrtf(fmaxf(A[i][i], 0.0f));
        float st = fmaxf(sv - inv_mu, 0.0f);   // tau = 1/mu
        f[i] = (sv > 1e-12f) ? st / sv : 0.0f;
      }
      for (int i = 0; i < KP; ++i)
        for (int j = 0; j < KP; ++j) {
          float s = 0.0f;
          if (i < KR && j < KR)
            for (int r = 0; r < KR; ++r) s += V[i][r] * f[r] * V[j][r];
          Mb[i * KP + j] = s;
        }
    }
    __syncthreads();

    // (10) QM = Q @ M -> Pbuf (P no longer needed this iteration)
    for (int ti = 0; ti < 4; ++ti) {
      const int m0 = (wave + ti * 8) * 16;
      v8f c = {0.f, 0.f, 0.f, 0.f, 0.f, 0.f, 0.f, 0.f};
      for (int k0 = 0; k0 < KP; k0 += 4) {
        v2f a;  a.x = ZQ[(m0 + l) * KP + k0 + 2 * half];
                a.y = ZQ[(m0 + l) * KP + k0 + 2 * half + 1];
        v2f bb; bb.x = Mb[(k0 + 2 * half) * KP + l];
                bb.y = Mb[(k0 + 2 * half + 1) * KP + l];
        c = wmma4(a, bb, c);
      }
      for (int g = 0; g < 8; ++g)
        Pbuf[(m0 + g + 8 * half) * KP + l] = c[g];
    }
    __syncthreads();

    // (11) L = QM @ Y_small  (512x16 * 16x512) -> global L
    for (int tt = wave; tt < 1024; tt += 8) {
      const int m0 = (tt >> 5) * 16;
      const int n0 = (tt & 31) * 16;
      v8f c = {0.f, 0.f, 0.f, 0.f, 0.f, 0.f, 0.f, 0.f};
      for (int k0 = 0; k0 < KP; k0 += 4) {
        v2f a;  a.x = Pbuf[(m0 + l) * KP + k0 + 2 * half];
                a.y = Pbuf[(m0 + l) * KP + k0 + 2 * half + 1];
        v2f bb; bb.x = Ys[(k0 + 2 * half) * WW + n0 + l];
                bb.y = Ys[(k0 + 2 * half + 1) * WW + n0 + l];
        c = wmma4(a, bb, c);
      }
      for (int g = 0; g < 8; ++g)
        Lg[(size_t)(m0 + g + 8 * half) * WW + n0 + l] = c[g];
    }
    __threadfence();
    __syncthreads();

    // (12)/(13) S = shr(D - L + Y/mu, lam/mu); Y += mu*(D-L-S); err
    float errsq = 0.0f;
    for (int i = t; i < ELEMS / 4; i += NTHR) {
      float4 d = D4[i], lv = L4[i], y = Y4[i];
      float4 s;
      s.x = shrinkf(d.x - lv.x + y.x * inv_mu, lam_mu);
      s.y = shrinkf(d.y - lv.y + y.y * inv_mu, lam_mu);
      s.z = shrinkf(d.z - lv.z + y.z * inv_mu, lam_mu);
      s.w = shrinkf(d.w - lv.w + y.w * inv_mu, lam_mu);
      S4[i] = s;
      float4 r;
      r.x = d.x - lv.x - s.x; r.y = d.y - lv.y - s.y;
      r.z = d.z - lv.z - s.z; r.w = d.w - lv.w - s.w;
      errsq += r.x * r.x + r.y * r.y + r.z * r.z + r.w * r.w;
      float4 yn;
      yn.x = y.x + mu * r.x; yn.y = y.y + mu * r.y;
      yn.z = y.z + mu * r.z; yn.w = y.w + mu * r.w;
      Y4[i] = yn;
    }
    __threadfence();
    float tot = block_reduce_sum(errsq, red);
    if (sqrtf(tot) * inv_normD < TOLF) break;   // uniform across block
  }
}

extern "C" void kernel_launch(void* const* d_in, const int* in_sizes, int n_in,
                              void* d_out, int out_size, void* d_ws, size_t ws_size,
                              hipStream_t stream) {
  (void)in_sizes; (void)n_in; (void)out_size; (void)ws_size;
  const float* D = (const float*)d_in[0];
  float* out = (float*)d_out;   // [L | S], each 64*512*512 floats
  float* ws  = (float*)d_ws;    // [Y | X], each 64*512*512 floats (128 MB)
  rpca_alm_kernel<<<NMAT, NTHR, 0, stream>>>(D, out, ws);
}